// CLIP_Embedding_35613868818658
// MI455X (gfx1250) — compile-verified
//
#include <hip/hip_runtime.h>

typedef __attribute__((ext_vector_type(2))) float v2f;
typedef __attribute__((ext_vector_type(8))) float v8f;

#define Bn   2
#define CINn 768
#define Ln   1024
#define DMn  512
#define DIn  1024
#define DSn  16
#define DTRn 32

// ---------------------------------------------------------------- CNN stem
__global__ void conv1d_k3(const float* __restrict__ x, const float* __restrict__ w,
                          const float* __restrict__ bias, float* __restrict__ y,
                          int Cin, int Cout) {
  int bc = blockIdx.x;
  int b = bc / Cout, co = bc % Cout;
  int l = blockIdx.y * blockDim.x + threadIdx.x;
  const float* xb = x + (size_t)b * Cin * Ln;
  const float* wc = w + (size_t)co * Cin * 3;
  float acc = bias[co];
  for (int ci = 0; ci < Cin; ++ci) {
    const float* xr = xb + (size_t)ci * Ln;
    float w0 = wc[ci * 3 + 0], w1 = wc[ci * 3 + 1], w2 = wc[ci * 3 + 2];
    float xm = (l > 0)      ? xr[l - 1] : 0.f;
    float xc = xr[l];
    float xp = (l < Ln - 1) ? xr[l + 1] : 0.f;
    acc = fmaf(w0, xm, acc);
    acc = fmaf(w1, xc, acc);
    acc = fmaf(w2, xp, acc);
  }
  y[((size_t)b * Cout + co) * Ln + l] = acc;
}

__global__ void gnorm_relu(const float* __restrict__ y, const float* __restrict__ g,
                           const float* __restrict__ bt, float* __restrict__ out, int C) {
  // one block per (b, group); 32 groups
  int b = blockIdx.x >> 5, grp = blockIdx.x & 31;
  int cpg = C >> 5;
  size_t base = ((size_t)b * C + (size_t)grp * cpg) * Ln;
  int n = cpg * Ln;
  float s = 0.f, s2 = 0.f;
  for (int i = threadIdx.x; i < n; i += blockDim.x) {
    float v = y[base + i]; s += v; s2 += v * v;
  }
  __shared__ float rs[256], rq[256];
  rs[threadIdx.x] = s; rq[threadIdx.x] = s2;
  __syncthreads();
  for (int st = 128; st > 0; st >>= 1) {
    if ((int)threadIdx.x < st) { rs[threadIdx.x] += rs[threadIdx.x + st]; rq[threadIdx.x] += rq[threadIdx.x + st]; }
    __syncthreads();
  }
  float mean = rs[0] / (float)n;
  float var  = rq[0] / (float)n - mean * mean;
  float rstd = rsqrtf(var + 1e-5f);
  for (int i = threadIdx.x; i < n; i += blockDim.x) {
    int c = grp * cpg + i / Ln;
    float v = (y[base + i] - mean) * rstd * g[c] + bt[c];
    out[base + i] = fmaxf(v, 0.f);
  }
}

// ---------------------------------------------------------------- LayerNorm (B,DM,L) -> (B*L, DM)
__global__ void layernorm_t(const float* __restrict__ h, const float* __restrict__ g,
                            const float* __restrict__ bt, float* __restrict__ out) {
  int m = blockIdx.x;               // b*L + l
  int b = m >> 10, l = m & (Ln - 1);
  const float* hb = h + (size_t)b * DMn * Ln + l;
  float s = 0.f, s2 = 0.f;
  for (int c = threadIdx.x; c < DMn; c += blockDim.x) {
    float v = hb[(size_t)c * Ln]; s += v; s2 += v * v;
  }
  __shared__ float rs[128], rq[128];
  rs[threadIdx.x] = s; rq[threadIdx.x] = s2;
  __syncthreads();
  for (int st = 64; st > 0; st >>= 1) {
    if ((int)threadIdx.x < st) { rs[threadIdx.x] += rs[threadIdx.x + st]; rq[threadIdx.x] += rq[threadIdx.x + st]; }
    __syncthreads();
  }
  float mean = rs[0] / (float)DMn;
  float var  = rq[0] / (float)DMn - mean * mean;
  float rstd = rsqrtf(var + 1e-5f);
  for (int c = threadIdx.x; c < DMn; c += blockDim.x) {
    float v = hb[(size_t)c * Ln];
    out[(size_t)m * DMn + c] = (v - mean) * rstd * g[c] + bt[c];
  }
}

// ---------------------------------------------------------------- WMMA fp32 GEMM: C = act(A*W^T + bias)
// A: (2048, lda) row-major; W: (N, ldw) row-major; one wave -> 16x64 tile.
// mode 0: C row-major (ldc). mode 1: C[(b*DMn+n)*Ln+l] = v + res[...]  (out_proj epilogue)
__global__ void wmma_gemm_f32(const float* __restrict__ A, int lda,
                              const float* __restrict__ W, int ldw,
                              const float* __restrict__ bias,
                              float* __restrict__ C, int ldc,
                              int K, int act, int mode,
                              const float* __restrict__ res) {
  const int lane = threadIdx.x;        // 32 threads per block (one wave)
  const int lid  = lane & 15;
  const int half = lane >> 4;
  const int mBase = blockIdx.x << 4;
  const int nBase = blockIdx.y << 6;

  v8f acc0 = {}, acc1 = {}, acc2 = {}, acc3 = {};
  const float* Ap = A + (size_t)(mBase + lid) * lda + (half << 1);
  const float* W0 = W + (size_t)(nBase +  0 + lid) * ldw + (half << 1);
  const float* W1 = W + (size_t)(nBase + 16 + lid) * ldw + (half << 1);
  const float* W2 = W + (size_t)(nBase + 32 + lid) * ldw + (half << 1);
  const float* W3 = W + (size_t)(nBase + 48 + lid) * ldw + (half << 1);

#pragma unroll 2
  for (int k = 0; k < K; k += 4) {
    v2f a  = *(const v2f*)(Ap + k);
    v2f b0 = *(const v2f*)(W0 + k);
    v2f b1 = *(const v2f*)(W1 + k);
    v2f b2 = *(const v2f*)(W2 + k);
    v2f b3 = *(const v2f*)(W3 + k);
    acc0 = __builtin_amdgcn_wmma_f32_16x16x4_f32(false, a, false, b0, (short)0, acc0, false, false);
    acc1 = __builtin_amdgcn_wmma_f32_16x16x4_f32(false, a, false, b1, (short)0, acc1, false, false);
    acc2 = __builtin_amdgcn_wmma_f32_16x16x4_f32(false, a, false, b2, (short)0, acc2, false, false);
    acc3 = __builtin_amdgcn_wmma_f32_16x16x4_f32(false, a, false, b3, (short)0, acc3, false, false);
  }

  v8f accs[4] = {acc0, acc1, acc2, acc3};
#pragma unroll
  for (int t = 0; t < 4; ++t) {
    int n = nBase + t * 16 + lid;
    float bv = bias ? bias[n] : 0.f;
#pragma unroll
    for (int r = 0; r < 8; ++r) {
      int m = mBase + r + (half << 3);
      float v = accs[t][r] + bv;
      if (act == 1) // softplus = max(x,0) + log1p(exp(-|x|))
        v = fmaxf(v, 0.f) + log1pf(expf(-fabsf(v)));
      if (mode == 0) {
        C[(size_t)m * ldc + n] = v;
      } else {
        int bb = m >> 10, l = m & (Ln - 1);
        size_t o = ((size_t)bb * DMn + n) * Ln + l;
        C[o] = v + res[o];
      }
    }
  }
}

// ---------------------------------------------------------------- depthwise causal conv k=4 + SiLU
// xz: (B*L, 2*DI) row-major (physical time). out: (B*L, DI) in LOGICAL time order.
__global__ void dwconv_silu(const float* __restrict__ xz, const float* __restrict__ w,
                            const float* __restrict__ bias, float* __restrict__ out,
                            int reverse) {
  size_t idx = (size_t)blockIdx.x * blockDim.x + threadIdx.x;  // over B*L*DI
  int c = (int)(idx % DIn);
  size_t m = idx / DIn;
  int t = (int)(m & (Ln - 1));
  int b = (int)(m >> 10);
  float acc = bias[c];
#pragma unroll
  for (int j = 0; j < 4; ++j) {
    int tt = t - 3 + j;
    if (tt >= 0) {
      int l = reverse ? (Ln - 1 - tt) : tt;
      acc = fmaf(w[c * 4 + j], xz[((size_t)b * Ln + l) * (2 * DIn) + c], acc);
    }
  }
  out[(size_t)m * DIn + c] = acc / (1.f + expf(-acc));   // silu
}

// ---------------------------------------------------------------- selective scan (logical time)
// u,delta: (B*L, DI); xdbl: (B*L, 64) with Bm=[32:48), Cm=[48:64). Writes at PHYSICAL index.
// B/C tile staged into LDS with gfx1250 async global->LDS DMA (ASYNCcnt tracked).
__global__ void scan_kernel(const float* __restrict__ u, const float* __restrict__ delta,
                            const float* __restrict__ xdbl, const float* __restrict__ Alog,
                            const float* __restrict__ Dp, float* __restrict__ ysum,
                            int reverse, int first) {
  int d = blockIdx.x * blockDim.x + threadIdx.x;   // channel
  int b = blockIdx.y;
  float Ac[DSn];
#pragma unroll
  for (int s = 0; s < DSn; ++s) Ac[s] = -expf(Alog[(size_t)d * DSn + s]);
  float Dd = Dp[d];
  float h[DSn];
#pragma unroll
  for (int s = 0; s < DSn; ++s) h[s] = 0.f;

  // [tt][0:16) = Bm, [tt][16:32) = Cm  (one dense 8KB tile per 64-step chunk)
  __shared__ __align__(16) float sBC[64][32];
  const float* gbase = xdbl + (size_t)b * Ln * 64 + 32;   // start of Bm columns

  for (int t0 = 0; t0 < Ln; t0 += 64) {
    __syncthreads();
    // 64 rows x 128B: 512 16-byte chunks, async-copied straight to LDS.
    // Low 32 bits of a flat shared pointer are the LDS byte offset (ISA 10.2).
#pragma unroll
    for (int q0 = 0; q0 < 512; q0 += 256) {
      int q = q0 + (int)threadIdx.x;
      int row = q >> 3, seg = q & 7;
      const float* gp = gbase + (size_t)(t0 + row) * 64 + seg * 4;
      unsigned int lp = (unsigned int)(size_t)&sBC[row][seg * 4];
      asm volatile("global_load_async_to_lds_b128 %0, %1, off"
                   :: "v"(lp), "v"(gp) : "memory");
    }
    if (t0 + 64 < Ln)
      __builtin_prefetch(&gbase[(size_t)(t0 + 64) * 64], 0, 0);
    asm volatile("s_wait_asynccnt 0x0" ::: "memory");
    __syncthreads();

    for (int tt = 0; tt < 64; ++tt) {
      int t = t0 + tt;
      size_t row = (size_t)b * Ln + t;
      float dlt = delta[row * DIn + d];
      float uu  = u[row * DIn + d];
      float du  = dlt * uu;
      float y = 0.f;
#pragma unroll
      for (int s = 0; s < DSn; ++s) {
        h[s] = expf(dlt * Ac[s]) * h[s] + du * sBC[tt][s];
        y = fmaf(h[s], sBC[tt][16 + s], y);
      }
      y = fmaf(uu, Dd, y);
      int l = reverse ? (Ln - 1 - t) : t;
      size_t o = ((size_t)b * Ln + l) * DIn + d;
      ysum[o] = first ? y : (ysum[o] + y);
    }
  }
}

// ---------------------------------------------------------------- gate: ysum *= silu(z)
__global__ void gate_combine(float* __restrict__ ysum, const float* __restrict__ xz) {
  size_t idx = (size_t)blockIdx.x * blockDim.x + threadIdx.x;  // over B*L*DI
  int c = (int)(idx % DIn);
  size_t m = idx / DIn;
  float z = xz[m * (2 * DIn) + DIn + c];
  ysum[idx] *= z / (1.f + expf(-z));
}

// ---------------------------------------------------------------- launcher
extern "C" void kernel_launch(void* const* d_in, const int* in_sizes, int n_in,
                              void* d_out, int out_size, void* d_ws, size_t ws_size,
                              hipStream_t stream) {
  (void)in_sizes; (void)n_in; (void)out_size; (void)ws_size;
  const float* x   = (const float*)d_in[0];
  const float* c1w = (const float*)d_in[1];  const float* c1b = (const float*)d_in[2];
  const float* g1g = (const float*)d_in[3];  const float* g1b = (const float*)d_in[4];
  const float* c2w = (const float*)d_in[5];  const float* c2b = (const float*)d_in[6];
  const float* g2g = (const float*)d_in[7];  const float* g2b = (const float*)d_in[8];
  const float* c3w = (const float*)d_in[9];  const float* c3b = (const float*)d_in[10];
  const float* g3g = (const float*)d_in[11]; const float* g3b = (const float*)d_in[12];
  const float* lng = (const float*)d_in[13]; const float* lnb = (const float*)d_in[14];
  const float* inW = (const float*)d_in[15]; const float* outW = (const float*)d_in[16];

  float* ws = (float*)d_ws;
  size_t off = 0;
  auto alloc = [&](size_t n) { float* p = ws + off; off += n; return p; };
  float* n1    = alloc((size_t)Bn * 128 * Ln);
  float* n2    = alloc((size_t)Bn * 256 * Ln);
  float* res   = alloc((size_t)Bn * 512 * Ln);
  float* convs = alloc((size_t)Bn * 512 * Ln);   // conv scratch; later reused as tnorm
  float* xz    = alloc((size_t)Bn * Ln * 2 * DIn);
  float* xconv = alloc((size_t)Bn * Ln * DIn);
  float* xdbl  = alloc((size_t)Bn * Ln * 64);
  float* delta = alloc((size_t)Bn * Ln * DIn);
  float* ysum  = alloc((size_t)Bn * Ln * DIn);
  float* tnorm = convs;

  const int BL = Bn * Ln;                 // 2048 rows for all GEMMs

  // --- stem ---
  conv1d_k3<<<dim3(Bn * 128, Ln / 256), 256, 0, stream>>>(x,  c1w, c1b, convs, CINn, 128);
  gnorm_relu<<<Bn * 32, 256, 0, stream>>>(convs, g1g, g1b, n1, 128);
  conv1d_k3<<<dim3(Bn * 256, Ln / 256), 256, 0, stream>>>(n1, c2w, c2b, convs, 128, 256);
  gnorm_relu<<<Bn * 32, 256, 0, stream>>>(convs, g2g, g2b, n2, 256);
  conv1d_k3<<<dim3(Bn * 512, Ln / 256), 256, 0, stream>>>(n2, c3w, c3b, convs, 256, 512);
  gnorm_relu<<<Bn * 32, 256, 0, stream>>>(convs, g3g, g3b, res, 512);

  // --- layernorm + in_proj ---
  layernorm_t<<<BL, 128, 0, stream>>>(res, lng, lnb, tnorm);
  // xz = tnorm @ in_proj_w^T : M=2048 K=512 N=2048
  wmma_gemm_f32<<<dim3(BL / 16, 2 * DIn / 64), 32, 0, stream>>>(
      tnorm, DMn, inW, DMn, nullptr, xz, 2 * DIn, DMn, /*act*/0, /*mode*/0, nullptr);

  // --- bidirectional mamba ---
  for (int dir = 0; dir < 2; ++dir) {
    const float* cvw = (const float*)d_in[17 + 7 * dir + 0];
    const float* cvb = (const float*)d_in[17 + 7 * dir + 1];
    const float* xpw = (const float*)d_in[17 + 7 * dir + 2];
    const float* dtw = (const float*)d_in[17 + 7 * dir + 3];
    const float* dtb = (const float*)d_in[17 + 7 * dir + 4];
    const float* Al  = (const float*)d_in[17 + 7 * dir + 5];
    const float* Dp  = (const float*)d_in[17 + 7 * dir + 6];

    dwconv_silu<<<(size_t)BL * DIn / 256, 256, 0, stream>>>(xz, cvw, cvb, xconv, dir);
    // xdbl = xconv @ xp_w^T : K=1024, N=64
    wmma_gemm_f32<<<dim3(BL / 16, 1), 32, 0, stream>>>(
        xconv, DIn, xpw, DIn, nullptr, xdbl, 64, DIn, 0, 0, nullptr);
    // delta = softplus(xdbl[:, :32] @ dt_w^T + dt_b) : K=32, N=1024
    wmma_gemm_f32<<<dim3(BL / 16, DIn / 64), 32, 0, stream>>>(
        xdbl, 64, dtw, DTRn, dtb, delta, DIn, DTRn, /*act*/1, 0, nullptr);
    scan_kernel<<<dim3(DIn / 256, Bn), 256, 0, stream>>>(
        xconv, delta, xdbl, Al, Dp, ysum, dir, dir == 0 ? 1 : 0);
  }

  // --- gate with silu(z), then out_proj with fused transpose+residual into d_out ---
  gate_combine<<<(size_t)BL * DIn / 256, 256, 0, stream>>>(ysum, xz);
  wmma_gemm_f32<<<dim3(BL / 16, DMn / 64), 32, 0, stream>>>(
      ysum, DIn, outW, DIn, nullptr, (float*)d_out, DMn, DIn, /*act*/0, /*mode*/1, res);
}